// SpikingVisionEmbedding_84902913507515
// MI455X (gfx1250) — compile-verified
//
#include <hip/hip_runtime.h>

typedef float v2f __attribute__((ext_vector_type(2)));
typedef float v8f __attribute__((ext_vector_type(8)));

#define D_MODEL   768
#define ROWS      6272              // B*N = 32*196
#define RB        32                // output rows per wave  (2 x 16 WMMA tiles)
#define CB        32                // output cols per wave  (2 x 16 WMMA tiles)
#define ROW_BLKS  (ROWS / RB)       // 196
#define COL_BLKS  (D_MODEL / CB)    // 24
#define WPB       8                 // waves per block
#define T_STEPS   16
#define V_TH      0.5f
#define BETA      0.5f

// One wave computes a 32x32 tile of z = X_patch * W^T + b + pos with 2x2
// register-blocked V_WMMA_F32_16X16X4_F32, then runs the 16-step LIF
// recurrence in registers and streams 16 spike tiles to HBM.
__global__ __launch_bounds__(256) void SpikingVisionEmbedding_kernel(
    const float* __restrict__ x,      // (32,3,224,224)
    const float* __restrict__ w,      // (768,3,16,16) row-major over K=768
    const float* __restrict__ bias,   // (768,)
    const float* __restrict__ pos,    // (1,196,768)
    float* __restrict__ out)          // (16,32,196,768)
{
    const int lane = threadIdx.x & 31;
    const int wave = threadIdx.x >> 5;
    const int tile = blockIdx.x * WPB + wave;

    const int rblk = tile / COL_BLKS;
    const int cblk = tile - rblk * COL_BLKS;
    const int row0 = rblk * RB;
    const int col0 = cblk * CB;

    const int l15 = lane & 15;
    const int hi  = lane >> 4;       // 0: lanes 0-15, 1: lanes 16-31

    // ---- A bases for the two 16-row tiles: row m -> x[b, :, h*16+p, w*16+q]
    long xbase[2];
    #pragma unroll
    for (int r = 0; r < 2; ++r) {
        const int m  = row0 + r * 16 + l15;
        const int bb = m / 196;
        const int n  = m - bb * 196;
        const int hh = n / 14;
        const int ww = n - hh * 14;
        xbase[r] = (long)bb * 3 * 50176 + (long)(hh * 16) * 224 + (long)(ww * 16);
    }

    // ---- B bases for the two 16-col tiles (w rows are contiguous over K)
    const long wbase0 = (long)(col0 + l15) * D_MODEL;
    const long wbase1 = wbase0 + 16 * D_MODEL;

    // ---- GEMM over K = 768 in steps of 4 (V_WMMA_F32_16X16X4_F32), 2x2 tiles
    v8f acc00 = {0.f,0.f,0.f,0.f,0.f,0.f,0.f,0.f};
    v8f acc01 = acc00, acc10 = acc00, acc11 = acc00;

    #pragma unroll 4
    for (int k0 = 0; k0 < D_MODEL; k0 += 4) {
        const int kk = k0 + 2 * hi;              // this lane-half's K pair
        const int ch = kk >> 8;                  // channel
        const int p  = (kk >> 4) & 15;           // pixel row in patch
        const int q  = kk & 15;                  // pixel col (stride-1 in x)
        const long xoff = (long)ch * 50176 + p * 224 + q;
        v2f a0 = *(const v2f*)(x + xbase[0] + xoff);
        v2f a1 = *(const v2f*)(x + xbase[1] + xoff);
        v2f b0 = *(const v2f*)(w + wbase0 + kk);
        v2f b1 = *(const v2f*)(w + wbase1 + kk);
        acc00 = __builtin_amdgcn_wmma_f32_16x16x4_f32(false, a0, false, b0, (short)0, acc00, false, false);
        acc01 = __builtin_amdgcn_wmma_f32_16x16x4_f32(false, a0, false, b1, (short)0, acc01, false, false);
        acc10 = __builtin_amdgcn_wmma_f32_16x16x4_f32(false, a1, false, b0, (short)0, acc10, false, false);
        acc11 = __builtin_amdgcn_wmma_f32_16x16x4_f32(false, a1, false, b1, (short)0, acc11, false, false);
    }

    // ---- bias + positional embedding (z overwrites accumulators)
    // C/D layout: register i, lanes 0-15 -> row +i, lanes 16-31 -> row +i+8
    const int n0 = row0 % 196;                   // scalar; rows span < 196
    const float bv0 = bias[col0 + l15];
    const float bv1 = bias[col0 + 16 + l15];

    float z[2][2][8];
    #pragma unroll
    for (int r = 0; r < 2; ++r) {
        #pragma unroll
        for (int c = 0; c < 2; ++c) {
            const v8f& a = r ? (c ? acc11 : acc10) : (c ? acc01 : acc00);
            const int colg = col0 + c * 16 + l15;
            const float bv = c ? bv1 : bv0;
            #pragma unroll
            for (int i = 0; i < 8; ++i) {
                const int roff = r * 16 + i + 8 * hi;
                int nn = n0 + roff;
                if (nn >= 196) nn -= 196;        // single wrap, no division
                z[r][c][i] = a[i] + bv + pos[nn * D_MODEL + colg];
            }
        }
    }

    // ---- LIF recurrence: t rolled, 32 elements unrolled; store addresses are
    //      per-lane base + compile-time immediate offsets.
    const long tstride = (long)ROWS * D_MODEL;   // elements per time step
    float* o = out + (long)(row0 + 8 * hi) * D_MODEL + col0 + l15;

    float vv[2][2][8];
    #pragma unroll
    for (int r = 0; r < 2; ++r)
        #pragma unroll
        for (int c = 0; c < 2; ++c)
            #pragma unroll
            for (int i = 0; i < 8; ++i) vv[r][c][i] = 0.f;

    #pragma unroll 1
    for (int t = 0; t < T_STEPS; ++t) {
        #pragma unroll
        for (int r = 0; r < 2; ++r) {
            #pragma unroll
            for (int c = 0; c < 2; ++c) {
                #pragma unroll
                for (int i = 0; i < 8; ++i) {
                    float v = BETA * vv[r][c][i] + z[r][c][i];
                    const float s = (v >= V_TH) ? 1.0f : 0.0f;
                    vv[r][c][i] = v - s * V_TH;  // soft reset
                    o[(r * 16 + i) * D_MODEL + c * 16] = s;  // imm offset
                }
            }
        }
        o += tstride;                            // one pointer bump per step
    }
}

extern "C" void kernel_launch(void* const* d_in, const int* in_sizes, int n_in,
                              void* d_out, int out_size, void* d_ws, size_t ws_size,
                              hipStream_t stream) {
    const float* x    = (const float*)d_in[0];   // (32,3,224,224)
    const float* w    = (const float*)d_in[1];   // (768,3,16,16)
    const float* bias = (const float*)d_in[2];   // (768,)
    const float* pos  = (const float*)d_in[3];   // (1,196,768)
    float* out = (float*)d_out;                  // (16,32,196,768)

    const int total_tiles = ROW_BLKS * COL_BLKS;             // 4704
    const int blocks = total_tiles / WPB;                    // 588
    SpikingVisionEmbedding_kernel<<<blocks, WPB * 32, 0, stream>>>(
        x, w, bias, pos, out);
}